// HGCN_22926535426452
// MI455X (gfx1250) — compile-verified
//
#include <hip/hip_runtime.h>
#include <math.h>

typedef __attribute__((ext_vector_type(2))) float v2f;
typedef __attribute__((ext_vector_type(8))) float v8f;

#define IN_DIM   128
#define EMB      10
#define OUTD     6
#define NUM_ET   32
#define NUM_BASE 8
#define HPAD     16   // h table row padding (cols 0-9 = h, 10-15 = h@root+bias)
#define H2PAD    8    // h2 table row padding (32B rows -> aligned float4 pairs)

// ---------------------------------------------------------------------------
// Tiny setup: B' = [w_emb | w_emb@root] (128x16), bias' (16), w = att@basis
// ---------------------------------------------------------------------------
__global__ void setup_kernel(const float* __restrict__ w_emb,
                             const float* __restrict__ b_emb,
                             const float* __restrict__ basis,
                             const float* __restrict__ att,
                             const float* __restrict__ root,
                             const float* __restrict__ bias,
                             float* __restrict__ bp,     // [128][16]
                             float* __restrict__ biasp,  // [16]
                             float* __restrict__ wrel)   // [32][10][6]
{
    int t = threadIdx.x;
    for (int i = t; i < IN_DIM * HPAD; i += blockDim.x) {
        int k = i >> 4, c = i & 15;
        float v;
        if (c < EMB) {
            v = w_emb[k * EMB + c];
        } else {
            int o = c - EMB;
            float s = 0.f;
            #pragma unroll
            for (int d = 0; d < EMB; ++d) s = fmaf(w_emb[k * EMB + d], root[d * OUTD + o], s);
            v = s;
        }
        bp[i] = v;
    }
    if (t < HPAD) {
        float v;
        if (t < EMB) {
            v = b_emb[t];
        } else {
            int o = t - EMB;
            float s = bias[o];
            #pragma unroll
            for (int d = 0; d < EMB; ++d) s = fmaf(b_emb[d], root[d * OUTD + o], s);
            v = s;
        }
        biasp[t] = v;
    }
    for (int i = t; i < NUM_ET * EMB * OUTD; i += blockDim.x) {
        int r = i / (EMB * OUTD), rem = i % (EMB * OUTD);
        float s = 0.f;
        #pragma unroll
        for (int b = 0; b < NUM_BASE; ++b)
            s = fmaf(att[r * NUM_BASE + b], basis[b * EMB * OUTD + rem], s);
        wrel[i] = s;
    }
}

// ---------------------------------------------------------------------------
// Encoder GEMM via V_WMMA_F32_16X16X4_F32:  [N x 128] @ [128 x 16]
// One 16-row tile per wave32; B' staged in LDS; 32 WMMA steps over K=128.
// ---------------------------------------------------------------------------
__global__ __launch_bounds__(256) void embed_wmma_kernel(
    const float* __restrict__ x,      // [N][128]
    const float* __restrict__ bp,     // [128][16]
    const float* __restrict__ biasp,  // [16]
    float* __restrict__ h,            // [N][16]
    int N)
{
    __shared__ float sbp[IN_DIM * HPAD];
    for (int i = threadIdx.x; i < IN_DIM * HPAD; i += blockDim.x) sbp[i] = bp[i];
    __syncthreads();

    int wave = blockIdx.x * (blockDim.x >> 5) + (threadIdx.x >> 5);
    int lane = threadIdx.x & 31;
    int row0 = wave * 16;
    if (row0 >= N) return;            // wave-uniform: EXEC stays all-ones for WMMA

    int half = lane >> 4;             // K-half select (A/B 32-bit layouts)
    int l16  = lane & 15;
    int arow = row0 + l16;
    if (arow >= N) arow = N - 1;      // clamp (N % 16 == 0 in practice)
    const float* xrow = x + (size_t)arow * IN_DIM;

    v8f c = {};
    #pragma unroll 4
    for (int k0 = 0; k0 < IN_DIM; k0 += 4) {
        int kb = k0 + 2 * half;
        v2f a, b;
        a.x = xrow[kb];
        a.y = xrow[kb + 1];
        b.x = sbp[kb * HPAD + l16];
        b.y = sbp[(kb + 1) * HPAD + l16];
        c = __builtin_amdgcn_wmma_f32_16x16x4_f32(false, a, false, b,
                                                  (short)0, c, false, false);
    }

    float badd = biasp[l16];          // col = l16
    #pragma unroll
    for (int v = 0; v < 8; ++v) {
        int m = row0 + v + 8 * half;  // C/D layout: VGPR v -> M = v (+8 for hi half)
        if (m < N) h[(size_t)m * HPAD + l16] = c[v] + badd;
    }
}

// ---------------------------------------------------------------------------
__global__ void zero_kernel(float* __restrict__ p, int n)
{
    int i = blockIdx.x * blockDim.x + threadIdx.x;
    if (i < n) p[i] = 0.0f;
}

// ---------------------------------------------------------------------------
// Per-edge message: msg = h[src] @ w[edge_type]; scatter-add into agg/deg.
// agg/deg tables are L2-resident (2.8 MB total).
// ---------------------------------------------------------------------------
__global__ __launch_bounds__(256) void msg_kernel(
    const int* __restrict__ src, const int* __restrict__ dst,
    const int* __restrict__ et,
    const float* __restrict__ h,      // [N][16]
    const float* __restrict__ wrel,   // [32][60]
    float* __restrict__ agg,          // [N][6]
    float* __restrict__ deg,          // [N]
    int E)
{
    __shared__ float sw[NUM_ET * EMB * OUTD];
    for (int i = threadIdx.x; i < NUM_ET * EMB * OUTD; i += blockDim.x) sw[i] = wrel[i];
    __syncthreads();

    for (int e = blockIdx.x * blockDim.x + threadIdx.x; e < E;
         e += gridDim.x * blockDim.x) {
        int s = src[e], d = dst[e], r = et[e];
        const float4* h4 = (const float4*)(h + (size_t)s * HPAD);
        float4 a0 = h4[0], a1 = h4[1];
        float2 a2 = ((const float2*)h4)[4];
        float hv[EMB] = { a0.x, a0.y, a0.z, a0.w, a1.x, a1.y, a1.z, a1.w, a2.x, a2.y };

        const float* wr = sw + r * (EMB * OUTD);
        float m[OUTD] = { 0.f, 0.f, 0.f, 0.f, 0.f, 0.f };
        #pragma unroll
        for (int dd = 0; dd < EMB; ++dd) {
            float xv = hv[dd];
            #pragma unroll
            for (int o = 0; o < OUTD; ++o) m[o] = fmaf(xv, wr[dd * OUTD + o], m[o]);
        }
        float* ag = agg + (size_t)d * OUTD;
        #pragma unroll
        for (int o = 0; o < OUTD; ++o) atomicAdd(ag + o, m[o]);
        atomicAdd(deg + d, 1.0f);
    }
}

// ---------------------------------------------------------------------------
// h2 = relu(agg / max(deg,1) + (h@root + bias))  -> padded [N][8]
// ---------------------------------------------------------------------------
__global__ void node_update_kernel(const float* __restrict__ h,
                                   const float* __restrict__ agg,
                                   const float* __restrict__ deg,
                                   float* __restrict__ h2, int N)
{
    int n = blockIdx.x * blockDim.x + threadIdx.x;
    if (n >= N) return;
    float dg = deg[n];
    float inv = 1.0f / (dg < 1.0f ? 1.0f : dg);
    #pragma unroll
    for (int o = 0; o < OUTD; ++o) {
        float v = fmaf(agg[(size_t)n * OUTD + o], inv, h[(size_t)n * HPAD + EMB + o]);
        h2[(size_t)n * H2PAD + o] = v > 0.f ? v : 0.f;
    }
    h2[(size_t)n * H2PAD + 6] = 0.f;
    h2[(size_t)n * H2PAD + 7] = 0.f;
}

// ---------------------------------------------------------------------------
// Decoder: sigmoid( sum_o h2[src][o] * rel[et][o] * h2[dst][o] )
// ---------------------------------------------------------------------------
__global__ __launch_bounds__(256) void score_kernel(
    const int* __restrict__ src, const int* __restrict__ dst,
    const int* __restrict__ et,
    const float* __restrict__ h2,      // [N][8]
    const float* __restrict__ rel_emb, // [32][6]
    float* __restrict__ out, int E)
{
    __shared__ float srel[NUM_ET * OUTD];
    for (int i = threadIdx.x; i < NUM_ET * OUTD; i += blockDim.x) srel[i] = rel_emb[i];
    __syncthreads();

    for (int e = blockIdx.x * blockDim.x + threadIdx.x; e < E;
         e += gridDim.x * blockDim.x) {
        int s = src[e], d = dst[e], r = et[e];
        const float4* hs = (const float4*)(h2 + (size_t)s * H2PAD);
        const float4* hd = (const float4*)(h2 + (size_t)d * H2PAD);
        float4 s0 = hs[0], s1 = hs[1];
        float4 d0 = hd[0], d1 = hd[1];
        const float* rr = srel + r * OUTD;
        float acc = s0.x * rr[0] * d0.x;
        acc = fmaf(s0.y * rr[1], d0.y, acc);
        acc = fmaf(s0.z * rr[2], d0.z, acc);
        acc = fmaf(s0.w * rr[3], d0.w, acc);
        acc = fmaf(s1.x * rr[4], d1.x, acc);
        acc = fmaf(s1.y * rr[5], d1.y, acc);
        out[e] = 1.0f / (1.0f + expf(-acc));
    }
}

// ---------------------------------------------------------------------------
extern "C" void kernel_launch(void* const* d_in, const int* in_sizes, int n_in,
                              void* d_out, int out_size, void* d_ws, size_t ws_size,
                              hipStream_t stream)
{
    const float* x       = (const float*)d_in[0];
    const float* w_emb   = (const float*)d_in[1];
    const float* b_emb   = (const float*)d_in[2];
    const float* basis   = (const float*)d_in[3];
    const float* att     = (const float*)d_in[4];
    const float* root    = (const float*)d_in[5];
    const float* bias    = (const float*)d_in[6];
    const float* rel_emb = (const float*)d_in[7];
    const int*   edge_index = (const int*)d_in[8];
    const int*   edge_type  = (const int*)d_in[9];
    // d_in[10] (edge_type_num) not needed: edge_type is per-edge already.

    int N = in_sizes[0] / IN_DIM;
    int E = in_sizes[8] / 2;
    const int* src = edge_index;
    const int* dst = edge_index + E;

    // Workspace carve-up (all 64B-aligned given N multiple of 16)
    char* ws = (char*)d_ws;
    float* bp    = (float*)(ws + 0);          // 128*16*4 = 8192 B
    float* biasp = (float*)(ws + 8192);       // 64 B (reserve 256)
    float* wrel  = (float*)(ws + 8448);       // 32*60*4 = 7680 B
    size_t off   = 16384;
    float* h     = (float*)(ws + off); off += (size_t)N * HPAD * sizeof(float);
    float* agg   = (float*)(ws + off); off += (size_t)N * OUTD * sizeof(float);
    float* deg   = (float*)(ws + off); off += (size_t)N * sizeof(float);
    float* h2    = (float*)(ws + off);
    float* out   = (float*)d_out;

    setup_kernel<<<1, 256, 0, stream>>>(w_emb, b_emb, basis, att, root, bias,
                                        bp, biasp, wrel);

    int nwaves  = (N + 15) / 16;
    int eblocks = (nwaves + 7) / 8;           // 8 waves (tiles) per 256-thread block
    embed_wmma_kernel<<<eblocks, 256, 0, stream>>>(x, bp, biasp, h, N);

    int nz = N * OUTD + N;                    // agg and deg are contiguous
    zero_kernel<<<(nz + 255) / 256, 256, 0, stream>>>(agg, nz);

    int mblocks = (E + 255) / 256;
    if (mblocks > 16384) mblocks = 16384;
    msg_kernel<<<mblocks, 256, 0, stream>>>(src, dst, edge_type, h, wrel, agg, deg, E);

    node_update_kernel<<<(N + 255) / 256, 256, 0, stream>>>(h, agg, deg, h2, N);

    score_kernel<<<mblocks, 256, 0, stream>>>(src, dst, edge_type, h2, rel_emb, out, E);
}